// COAConv_6167573037222
// MI455X (gfx1250) — compile-verified
//
#include <hip/hip_runtime.h>
#include <hip/hip_bf16.h>
#include <math.h>

// Problem constants (match reference)
#define NN   40000
#define EE   640000
#define HH   4
#define HS_  16
#define NT_  4
#define ET_  8

typedef __attribute__((ext_vector_type(2))) float v2f;
typedef __attribute__((ext_vector_type(8))) float v8f;
typedef int v4i __attribute__((vector_size(16)));

// ---------------------------------------------------------------------------
// Optional CDNA5 async global->LDS staging (guarded: falls back to plain copy)
// Builtin signature (from probe diagnostic): param0 = v4i addrspace(1)*,
// param1 = v4i addrspace(3)*, then imm offset + cpol.
// ---------------------------------------------------------------------------
#if __has_builtin(__builtin_amdgcn_global_load_async_to_lds_b128) && \
    __has_builtin(__builtin_amdgcn_s_wait_asynccnt)
#define HAVE_ASYNC_LDS 1
#else
#define HAVE_ASYNC_LDS 0
#endif

__device__ __forceinline__ void stage_lds(float* __restrict__ sdst,
                                          const float* __restrict__ gsrc,
                                          int ndw, int tid, int nth) {
#if HAVE_ASYNC_LDS
  for (int i = tid * 4; i < ndw; i += nth * 4)
    __builtin_amdgcn_global_load_async_to_lds_b128(
        (__attribute__((address_space(1))) v4i*)(gsrc + i),
        (__attribute__((address_space(3))) v4i*)(sdst + i), 0, 0);
#else
  for (int i = tid * 4; i < ndw; i += nth * 4)
    *(float4*)(sdst + i) = *(const float4*)(gsrc + i);
#endif
}

__device__ __forceinline__ void wait_stage() {
#if HAVE_ASYNC_LDS
  __builtin_amdgcn_s_wait_asynccnt(0);
#endif
}

// ---------------------------------------------------------------------------
// WMMA helper: D = A(16x4,f32) * B(4x16,f32) + C(16x16,f32)
// A layout: lane m=lane&15; VGPR0 = A[m][k0 + (lane>=16?2:0)], VGPR1 = +1
// B layout: lane n=lane&15; VGPR0 = B[k0 + (lane>=16?2:0)][n], VGPR1 = +1
// C layout: VGPR r -> row M = r + (lane>=16 ? 8 : 0), col N = lane&15
// ---------------------------------------------------------------------------
__device__ __forceinline__ v8f wmma4(v2f a, v2f b, v8f c) {
  return __builtin_amdgcn_wmma_f32_16x16x4_f32(
      /*neg_a=*/false, a, /*neg_b=*/false, b,
      /*c_mod=*/(short)0, c, /*reuse_a=*/false, /*reuse_b=*/false);
}

// ---------------------------------------------------------------------------
// Init: zero hbuf, amax=-inf, denom=0
// ---------------------------------------------------------------------------
__global__ void init_kernel(float* __restrict__ hbuf, float* __restrict__ amax,
                            float* __restrict__ denom) {
  int tid = blockIdx.x * blockDim.x + threadIdx.x;
  if (tid < NN * 64) hbuf[tid] = 0.0f;
  if (tid < NN * HH) { amax[tid] = -3.0e38f; denom[tid] = 0.0f; }
}

// ---------------------------------------------------------------------------
// Typed linear: out[16-node tile] = sum_t (mask_t . x) @ W[t]   (64 -> 64)
// 256-thread block = 8 waves = 8 node tiles; W (all 4 types, 64KB) staged in
// LDS once per block, B-fragments then come from ds_load.
// ---------------------------------------------------------------------------
#define TP_WAVES 8
__global__ __launch_bounds__(256)
void typed_proj_kernel(const float* __restrict__ xin, const int* __restrict__ ntype,
                       const float* __restrict__ W, float* __restrict__ out) {
  __shared__ float sW[NT_ * 64 * 64];        // 64 KB
  __shared__ float sx[TP_WAVES * 16 * 64];   // 32 KB
  __shared__ int   snt[TP_WAVES * 16];

  const int tid  = threadIdx.x;
  const int wid  = tid >> 5;
  const int lane = tid & 31;
  int tile = blockIdx.x * TP_WAVES + wid;
  if (tile >= NN / 16) tile = 0;             // tail waves redo tile 0 (benign)
  const int g0 = tile * 16;

  __builtin_prefetch(W, 0, 1);

  // Stage weights (cooperative) and per-wave x tile into LDS.
  stage_lds(sW, W, NT_ * 64 * 64, tid, 256);
  stage_lds(sx + wid * 1024, xin + g0 * 64, 1024, lane, 32);
  if (lane < 16) snt[wid * 16 + lane] = ntype[g0 + lane];
  wait_stage();
  __syncthreads();

  const int m       = lane & 15;
  const int koffadd = (lane >> 4) << 1;      // 0 or 2
  const int n       = lane & 15;
  const int half8   = (lane >> 4) << 3;
  const float* xw   = sx + wid * 1024;

  v2f afrag[16];
#pragma unroll
  for (int c = 0; c < 16; ++c) {
    const int kk = 4 * c + koffadd;
    afrag[c].x = xw[m * 64 + kk];
    afrag[c].y = xw[m * 64 + kk + 1];
  }
  const int myt = snt[wid * 16 + m];

#pragma unroll
  for (int n0 = 0; n0 < 64; n0 += 16) {
    v8f acc = {};
    for (int t = 0; t < NT_; ++t) {
      if (__ballot(myt == t) == 0) continue;   // wave-uniform skip
      const float msk = (myt == t) ? 1.0f : 0.0f;
      const float* Bt = sW + t * 4096 + n0 + n;
#pragma unroll
      for (int c = 0; c < 16; ++c) {
        const int kb = 4 * c + koffadd;
        v2f a = afrag[c] * msk;
        v2f b;
        b.x = Bt[kb * 64];
        b.y = Bt[(kb + 1) * 64];
        acc = wmma4(a, b, acc);
      }
    }
#pragma unroll
    for (int r = 0; r < 8; ++r) {
      const int M = r + half8;
      out[(g0 + M) * 64 + n0 + n] = acc[r];
    }
  }
}

// ---------------------------------------------------------------------------
// G[h][et] = R @ R^T  for R = rel_att[h][et] (16x16).  32 waves total.
// ---------------------------------------------------------------------------
__global__ __launch_bounds__(32)
void relG_kernel(const float* __restrict__ rel_att, float* __restrict__ G) {
  const int w    = blockIdx.x;               // h*ET + et
  const float* R = rel_att + w * 256;
  const int lane = threadIdx.x;
#pragma unroll
  for (int p = 0; p < 8; ++p) {
    const int idx = lane * 8 + p;
    const int i = idx >> 4, j = idx & 15;
    float s = 0.0f;
#pragma unroll
    for (int o = 0; o < 16; ++o) s += R[i * 16 + o] * R[j * 16 + o];
    G[w * 256 + idx] = s;
  }
}

// ---------------------------------------------------------------------------
// Edge logits: a[e,h] = ke G(h,et) qe^T * pri / 4 ; atomicMax into amax[dst,h]
// 256-thread block = 8 waves = 8 edge tiles of one head; G[h] staged in LDS.
// ---------------------------------------------------------------------------
#define EW 8
__global__ __launch_bounds__(256)
void edge_logits_kernel(const float* __restrict__ k, const float* __restrict__ q,
                        const int* __restrict__ src, const int* __restrict__ dst,
                        const int* __restrict__ etype, const float* __restrict__ G,
                        const float* __restrict__ rel_pri,
                        float* __restrict__ a_out, float* __restrict__ amax) {
  __shared__ float sG[ET_ * 256];            // 8 KB
  __shared__ float ske[EW][256], sqe[EW][256];
  __shared__ int   ssrc[EW][16], sdst[EW][16], setp[EW][16];

  const int tid  = threadIdx.x;
  const int wid  = tid >> 5;
  const int lane = tid & 31;
  const int h    = blockIdx.y;
  const int e0   = (blockIdx.x * EW + wid) * 16;

  stage_lds(sG, G + h * ET_ * 256, ET_ * 256, tid, 256);
  if (lane < 16) {
    ssrc[wid][lane] = src[e0 + lane];
    sdst[wid][lane] = dst[e0 + lane];
    setp[wid][lane] = etype[e0 + lane];
  }
  __syncthreads();                           // indices visible

  // Vectorized gathers: each row is 16 contiguous floats (one head slice).
  for (int u = lane; u < 64; u += 32) {
    const int row = u >> 2, seg = u & 3;
    *(float4*)&ske[wid][row * 16 + seg * 4] =
        *(const float4*)&k[ssrc[wid][row] * 64 + h * 16 + seg * 4];
    *(float4*)&sqe[wid][row * 16 + seg * 4] =
        *(const float4*)&q[sdst[wid][row] * 64 + h * 16 + seg * 4];
  }
  wait_stage();
  __syncthreads();

  const int m = lane & 15;
  const int koffadd = (lane >> 4) << 1;
  const int n = lane & 15;
  const int half8 = (lane >> 4) << 3;

  v2f afrag[4];
#pragma unroll
  for (int c = 0; c < 4; ++c) {
    const int kk = 4 * c + koffadd;
    afrag[c].x = ske[wid][m * 16 + kk];
    afrag[c].y = ske[wid][m * 16 + kk + 1];
  }
  const int myt = setp[wid][m];

  v8f acc = {};
  for (int t = 0; t < ET_; ++t) {
    if (__ballot(myt == t) == 0) continue;   // wave-uniform skip
    const float msk = (myt == t) ? 1.0f : 0.0f;
    const float* Gt = sG + t * 256 + n;
#pragma unroll
    for (int c = 0; c < 4; ++c) {
      const int kb = 4 * c + koffadd;
      v2f a = afrag[c] * msk;
      v2f b;
      b.x = Gt[kb * 16];
      b.y = Gt[(kb + 1) * 16];
      acc = wmma4(a, b, acc);
    }
  }

  // Row-wise dot with qe: reduce over N (16 lanes per half-wave)
#pragma unroll
  for (int r = 0; r < 8; ++r) {
    const int M = r + half8;
    float p = acc[r] * sqe[wid][M * 16 + n];
    p += __shfl_xor(p, 1, 32);
    p += __shfl_xor(p, 2, 32);
    p += __shfl_xor(p, 4, 32);
    p += __shfl_xor(p, 8, 32);
    if (n == 0) {
      const float val = p * rel_pri[h * ET_ + setp[wid][M]] * 0.25f;  // /sqrt(16)
      a_out[(e0 + M) * HH + h] = val;
      atomicMax(&amax[sdst[wid][M] * HH + h], val);
    }
  }
}

// ---------------------------------------------------------------------------
// Softmax numerator + denominator accumulation
// ---------------------------------------------------------------------------
__global__ void edge_softmax_kernel(float* __restrict__ a, const int* __restrict__ dst,
                                    const float* __restrict__ amax,
                                    float* __restrict__ denom) {
  const int tid = blockIdx.x * blockDim.x + threadIdx.x;
  if (tid >= EE * HH) return;
  const int e = tid >> 2;
  const int h = tid & 3;
  const int d = dst[e];
  const float ex = expf(a[tid] - amax[d * HH + h]);
  a[tid] = ex;
  atomicAdd(&denom[d * HH + h], ex);
}

// ---------------------------------------------------------------------------
// Aggregate: msg = (mask_t . ve) @ rel_msg[h][t]; hbuf[dst] += alpha * msg
// ---------------------------------------------------------------------------
__global__ __launch_bounds__(256)
void edge_aggregate_kernel(const float* __restrict__ v, const int* __restrict__ src,
                           const int* __restrict__ dst, const int* __restrict__ etype,
                           const float* __restrict__ rel_msg,
                           const float* __restrict__ aex, const float* __restrict__ denom,
                           float* __restrict__ hbuf) {
  __shared__ float sR[ET_ * 256];            // rel_msg[h] tables, 8 KB
  __shared__ float sve[EW][256];
  __shared__ float salpha[EW][16];
  __shared__ int   ssrc[EW][16], sdst[EW][16], setp[EW][16];

  const int tid  = threadIdx.x;
  const int wid  = tid >> 5;
  const int lane = tid & 31;
  const int h    = blockIdx.y;
  const int e0   = (blockIdx.x * EW + wid) * 16;

  stage_lds(sR, rel_msg + h * ET_ * 256, ET_ * 256, tid, 256);
  if (lane < 16) {
    ssrc[wid][lane] = src[e0 + lane];
    sdst[wid][lane] = dst[e0 + lane];
    setp[wid][lane] = etype[e0 + lane];
  }
  __syncthreads();

  for (int u = lane; u < 64; u += 32) {
    const int row = u >> 2, seg = u & 3;
    *(float4*)&sve[wid][row * 16 + seg * 4] =
        *(const float4*)&v[ssrc[wid][row] * 64 + h * 16 + seg * 4];
  }
  if (lane < 16) {
    salpha[wid][lane] = aex[(e0 + lane) * HH + h] / denom[sdst[wid][lane] * HH + h];
  }
  wait_stage();
  __syncthreads();

  const int m = lane & 15;
  const int koffadd = (lane >> 4) << 1;
  const int n = lane & 15;
  const int half8 = (lane >> 4) << 3;

  v2f afrag[4];
#pragma unroll
  for (int c = 0; c < 4; ++c) {
    const int kk = 4 * c + koffadd;
    afrag[c].x = sve[wid][m * 16 + kk];
    afrag[c].y = sve[wid][m * 16 + kk + 1];
  }
  const int myt = setp[wid][m];

  v8f acc = {};
  for (int t = 0; t < ET_; ++t) {
    if (__ballot(myt == t) == 0) continue;   // wave-uniform skip
    const float msk = (myt == t) ? 1.0f : 0.0f;
    const float* Rt = sR + t * 256 + n;
#pragma unroll
    for (int c = 0; c < 4; ++c) {
      const int kb = 4 * c + koffadd;
      v2f a = afrag[c] * msk;
      v2f b;
      b.x = Rt[kb * 16];
      b.y = Rt[(kb + 1) * 16];
      acc = wmma4(a, b, acc);
    }
  }

#pragma unroll
  for (int r = 0; r < 8; ++r) {
    const int M = r + half8;
    atomicAdd(&hbuf[sdst[wid][M] * 64 + h * 16 + n], acc[r] * salpha[wid][M]);
  }
}

// ---------------------------------------------------------------------------
// Host launcher
// ---------------------------------------------------------------------------
extern "C" void kernel_launch(void* const* d_in, const int* in_sizes, int n_in,
                              void* d_out, int out_size, void* d_ws, size_t ws_size,
                              hipStream_t stream) {
  const float* x       = (const float*)d_in[0];
  const int*   ntype   = (const int*)  d_in[1];
  const int*   etype   = (const int*)  d_in[2];
  const int*   src     = (const int*)  d_in[3];
  const int*   dst     = (const int*)  d_in[4];
  const float* Wk      = (const float*)d_in[5];
  const float* Wq      = (const float*)d_in[6];
  const float* Wv      = (const float*)d_in[7];
  const float* Wa      = (const float*)d_in[8];
  const float* rel_att = (const float*)d_in[9];
  const float* rel_msg = (const float*)d_in[10];
  const float* rel_pri = (const float*)d_in[11];

  float* ws    = (float*)d_ws;
  float* k     = ws;                                   // N*64
  float* q     = ws + (size_t)NN * 64;                 // N*64
  float* v     = ws + (size_t)2 * NN * 64;             // N*64
  float* hbuf  = ws + (size_t)3 * NN * 64;             // N*64
  float* G     = ws + (size_t)4 * NN * 64;             // H*ET*256 = 8192
  float* a     = G + HH * ET_ * 256;                   // E*H
  float* amax  = a + (size_t)EE * HH;                  // N*H
  float* denom = amax + (size_t)NN * HH;               // N*H

  const int nNodeTiles = NN / 16;                      // 2500
  const int nEdgeTiles = EE / 16;                      // 40000
  const int nProjBlks  = (nNodeTiles + TP_WAVES - 1) / TP_WAVES;  // 313
  const int nEdgeBlks  = nEdgeTiles / EW;              // 5000

  init_kernel<<<(NN * 64 + 255) / 256, 256, 0, stream>>>(hbuf, amax, denom);

  typed_proj_kernel<<<nProjBlks, 256, 0, stream>>>(x, ntype, Wk, k);
  typed_proj_kernel<<<nProjBlks, 256, 0, stream>>>(x, ntype, Wq, q);
  typed_proj_kernel<<<nProjBlks, 256, 0, stream>>>(x, ntype, Wv, v);

  relG_kernel<<<HH * ET_, 32, 0, stream>>>(rel_att, G);

  edge_logits_kernel<<<dim3(nEdgeBlks, HH), 256, 0, stream>>>(
      k, q, src, dst, etype, G, rel_pri, a, amax);

  edge_softmax_kernel<<<(EE * HH + 255) / 256, 256, 0, stream>>>(a, dst, amax, denom);

  edge_aggregate_kernel<<<dim3(nEdgeBlks, HH), 256, 0, stream>>>(
      v, src, dst, etype, rel_msg, a, denom, hbuf);

  typed_proj_kernel<<<nProjBlks, 256, 0, stream>>>(hbuf, ntype, Wa, (float*)d_out);
}